// ToRGB_43980465111655
// MI455X (gfx1250) — compile-verified
//
#include <hip/hip_runtime.h>

typedef __attribute__((ext_vector_type(2))) float v2f;
typedef __attribute__((ext_vector_type(4))) float v4f;
typedef __attribute__((ext_vector_type(8))) float v8f;

#define IN_CH 512
#define W_DIM 512
#define OUT_CH 3
#define PLANE 16384   // 128*128
#define CLAMP_V 256.0f

// ---------------------------------------------------------------------------
// Kernel 1: styles GEMM on the WMMA path (V_WMMA_F32_16X16X4_F32) + fused
// coefficient precompute.  One wave (32 threads) per 16-wide output tile.
//   styles[b,i] = lrelu(sum_k w[b,k]*affine_w[i,k]/sqrt(512) + affine_b[i]) * sqrt(2)
//   coef[b,o,i] = weight[o,i] * (1/512) * styles[b,i]
// ---------------------------------------------------------------------------
__global__ void __launch_bounds__(32) torgb_affine_wmma(
    const float* __restrict__ w,         // [8, 512]
    const float* __restrict__ affine_w,  // [512, 512]
    const float* __restrict__ affine_b,  // [512]
    const float* __restrict__ weight,    // [3, 512] (1x1 conv weights)
    float* __restrict__ coef)            // [8, 3, 512] output
{
  const int lane = threadIdx.x;
  const int i0   = blockIdx.x * 16;          // output-channel tile base
  const int r    = lane & 15;                // A: M row / B: N column
  const int kb   = (lane < 16) ? 0 : 2;      // K sub-offset per half-wave
  const float aw_gain = 0.044194173824159216f;   // 1/sqrt(512)
  const float amask   = (r < 8) ? 1.0f : 0.0f;   // pad batch rows 8..15 to 0

  const float* wrow = w        + (size_t)(r & 7) * W_DIM + kb;  // branchless, in-bounds
  const float* brow = affine_w + (size_t)(i0 + r) * W_DIM + kb;

  v8f c = {};
  for (int k = 0; k < W_DIM; k += 4) {
    v2f a, b;
    a.x = wrow[k]     * amask;   // A: lane<16 -> K=k,k+1 ; lane>=16 -> K=k+2,k+3
    a.y = wrow[k + 1] * amask;
    b.x = brow[k]     * aw_gain; // B mirrors A's K striping, N = lane&15
    b.y = brow[k + 1] * aw_gain;
    // 8 args: (neg_a, A, neg_b, B, c_mod, C, reuse_a, reuse_b)
    c = __builtin_amdgcn_wmma_f32_16x16x4_f32(false, a, false, b,
                                              (short)0, c, false, false);
  }

  // D layout: lanes 0..15, VGPR v -> D[M=v, N=lane] = styles_raw[batch=v, i0+lane]
  if (lane < 16) {
    const int i   = i0 + lane;
    const float ab = affine_b[i];
    const float wg = 1.0f / (float)IN_CH;
    float wm[OUT_CH];
#pragma unroll
    for (int o = 0; o < OUT_CH; ++o) wm[o] = weight[o * IN_CH + i] * wg;
#pragma unroll
    for (int v = 0; v < 8; ++v) {            // batch = v
      float s = c[v] + ab;
      s = (s < 0.0f) ? 0.2f * s : s;         // leaky_relu(0.2)
      s *= 1.41421356237309515f;             // gain sqrt(2)
#pragma unroll
      for (int o = 0; o < OUT_CH; ++o)
        coef[((size_t)v * OUT_CH + o) * IN_CH + i] = wm[o] * s;
    }
  }
}

// ---------------------------------------------------------------------------
// Separable FIR upsample tap helper: k1 = [0.25, 0.75, 0.75, 0.25] per axis,
// zero-insert x2 upsampling (gain 4 folded into k1 outer product).
// ---------------------------------------------------------------------------
__device__ __forceinline__ float skip_tap(const float* __restrict__ sp,
                                          int sh, int sw) {
  if ((unsigned)sh > 63u || (unsigned)sw > 63u) return 0.0f;
  return sp[sh * 64 + sw];
}

__device__ __forceinline__ float upsample_at(const float* __restrict__ sp,
                                             int h, int w) {
  int sh0, sh1, sw0, sw1;
  float kh0, kh1, kw0, kw1;
  if (h & 1) { sh0 = (h - 1) >> 1; sh1 = sh0 + 1; kh0 = 0.75f; kh1 = 0.25f; }
  else       { sh1 = h >> 1;       sh0 = sh1 - 1; kh0 = 0.25f; kh1 = 0.75f; }
  if (w & 1) { sw0 = (w - 1) >> 1; sw1 = sw0 + 1; kw0 = 0.75f; kw1 = 0.25f; }
  else       { sw1 = w >> 1;       sw0 = sw1 - 1; kw0 = 0.25f; kw1 = 0.75f; }
  float r0 = kw0 * skip_tap(sp, sh0, sw0) + kw1 * skip_tap(sp, sh0, sw1);
  float r1 = kw0 * skip_tap(sp, sh1, sw0) + kw1 * skip_tap(sp, sh1, sw1);
  return kh0 * r0 + kh1 * r1;
}

// ---------------------------------------------------------------------------
// Kernel 2: bandwidth pass. y[b,o,p] = clamp(sum_i coef[b,o,i]*x[b,i,p] + bias)
//           + upsample2d(skip).  Each thread owns 4 consecutive pixels
//           (v4f / global_load_b128 per channel), coef table in LDS.
// x (256 MB, zero reuse, > 192 MB L2) is streamed with non-temporal hints;
// output is stored non-temporally (written once, never re-read).
// grid = (16, 8): blockIdx.y = batch, 16 blocks x 256 threads cover 128*128/4.
// ---------------------------------------------------------------------------
__global__ void __launch_bounds__(256) torgb_main(
    const float* __restrict__ x,     // [8, 512, 128, 128]
    const float* __restrict__ skip,  // [8, 3, 64, 64]
    const float* __restrict__ coef,  // [8, 3, 512]
    const float* __restrict__ bias,  // [3]
    float* __restrict__ out)         // [8, 3, 128, 128]
{
  __shared__ float sc[OUT_CH * IN_CH];   // 6 KB
  const int b   = blockIdx.y;
  const int tid = threadIdx.x;

  for (int idx = tid; idx < OUT_CH * IN_CH; idx += 256)
    sc[idx] = coef[(size_t)b * OUT_CH * IN_CH + idx];
  __syncthreads();

  const int p4   = blockIdx.x * 256 + tid;  // v4f group within batch plane
  const int pix0 = p4 * 4;
  const int h    = pix0 >> 7;
  const int w0   = pix0 & 127;              // 4 pixels share one row

  const v4f* __restrict__ xb =
      (const v4f*)x + (size_t)b * IN_CH * (PLANE / 4) + p4;

  v4f a0 = {0.f, 0.f, 0.f, 0.f};
  v4f a1 = {0.f, 0.f, 0.f, 0.f};
  v4f a2 = {0.f, 0.f, 0.f, 0.f};

  // 8 independent non-temporal b128 loads in flight per iteration group:
  // need ~32 outstanding loads/wave to cover BW*latency with 1024 waves.
#pragma unroll 8
  for (int i = 0; i < IN_CH; ++i) {
    v4f xv = __builtin_nontemporal_load(&xb[(size_t)i * (PLANE / 4)]);
    float c0 = sc[i];                               // LDS broadcasts
    float c1 = sc[IN_CH + i];
    float c2 = sc[2 * IN_CH + i];
    a0.x = fmaf(c0, xv.x, a0.x); a0.y = fmaf(c0, xv.y, a0.y);
    a0.z = fmaf(c0, xv.z, a0.z); a0.w = fmaf(c0, xv.w, a0.w);
    a1.x = fmaf(c1, xv.x, a1.x); a1.y = fmaf(c1, xv.y, a1.y);
    a1.z = fmaf(c1, xv.z, a1.z); a1.w = fmaf(c1, xv.w, a1.w);
    a2.x = fmaf(c2, xv.x, a2.x); a2.y = fmaf(c2, xv.y, a2.y);
    a2.z = fmaf(c2, xv.z, a2.z); a2.w = fmaf(c2, xv.w, a2.w);
  }

  v4f acc[OUT_CH] = {a0, a1, a2};
  const float* sb = skip + (size_t)b * OUT_CH * 4096;   // 64*64 planes
  v4f* __restrict__ out4 = (v4f*)out + (size_t)b * OUT_CH * (PLANE / 4);

#pragma unroll
  for (int o = 0; o < OUT_CH; ++o) {
    const float bo = bias[o];
    const float* sp = sb + o * 4096;
    v4f res;
#pragma unroll
    for (int j = 0; j < 4; ++j) {
      float v = acc[o][j] + bo;
      v = fminf(fmaxf(v, -CLAMP_V), CLAMP_V);
      res[j] = v + upsample_at(sp, h, w0 + j);
    }
    __builtin_nontemporal_store(res, &out4[(size_t)o * (PLANE / 4) + p4]);
  }
}

// ---------------------------------------------------------------------------
extern "C" void kernel_launch(void* const* d_in, const int* in_sizes, int n_in,
                              void* d_out, int out_size, void* d_ws,
                              size_t ws_size, hipStream_t stream) {
  const float* x        = (const float*)d_in[0];  // [8,512,128,128]
  const float* w        = (const float*)d_in[1];  // [8,512]
  const float* skip     = (const float*)d_in[2];  // [8,3,64,64]
  const float* affine_w = (const float*)d_in[3];  // [512,512]
  const float* affine_b = (const float*)d_in[4];  // [512]
  const float* weight   = (const float*)d_in[5];  // [3,512,1,1]
  const float* bias     = (const float*)d_in[6];  // [3]
  float* out  = (float*)d_out;
  float* coef = (float*)d_ws;                     // 8*3*512 floats = 48 KB

  // 32 tiles of 16 output channels; one wave each.
  torgb_affine_wmma<<<dim3(32), dim3(32), 0, stream>>>(w, affine_w, affine_b,
                                                       weight, coef);
  // Bandwidth pass: 16 blocks per batch x 8 batches, 256 threads.
  torgb_main<<<dim3(16, 8), dim3(256), 0, stream>>>(x, skip, coef, bias, out);
}